// EdgePooling_34479997452436
// MI455X (gfx1250) — compile-verified
//
#include <hip/hip_runtime.h>
#include <stdint.h>

typedef __attribute__((ext_vector_type(2))) float v2f;
typedef __attribute__((ext_vector_type(8))) float v8f;
typedef __attribute__((ext_vector_type(4))) unsigned v4u;
typedef __attribute__((ext_vector_type(8))) int v8i;
typedef __attribute__((ext_vector_type(4))) int v4i;
typedef unsigned long long u64;

#define THREADS 256

// ---------- helpers ----------
__device__ __forceinline__ unsigned f32_ord(float f) {
  unsigned u = __float_as_uint(f);
  return (u & 0x80000000u) ? ~u : (u | 0x80000000u);
}
__device__ __forceinline__ float ord_f32(unsigned u) {
  return __uint_as_float((u & 0x80000000u) ? (u & 0x7FFFFFFFu) : ~u);
}

__global__ void k_fill_f32(float* __restrict__ p, float v, int n) {
  int i = blockIdx.x * blockDim.x + threadIdx.x;
  if (i < n) p[i] = v;
}
__global__ void k_fill_i32(int* __restrict__ p, int v, int n) {
  int i = blockIdx.x * blockDim.x + threadIdx.x;
  if (i < n) p[i] = v;
}

// ---------- node projection: p1 = x @ W[:C], p2 = x @ W[C:] ----------
// TDM loads each wave's 16xC f32 tile into LDS (row stride C+1 via D# padding,
// de-conflicting banks), WMMA f32 16x16x4 consumes it from LDS.
// B matrix: N=0 -> W[:C], N=1 -> W[C:], N>=2 zero (branchless mask).
#define CC 128           // C is fixed at 128 for this workload
#define CSTRIDE 129      // C + 1 dword LDS pad per row

__global__ __launch_bounds__(256) void k_proj(const float* __restrict__ x,
                                              const float* __restrict__ W,
                                              float* __restrict__ p1,
                                              float* __restrict__ p2,
                                              int N) {
  __shared__ float ldsX[8][16 * CSTRIDE];  // 8 waves x (16 rows x 129)
  __shared__ float ldsW[2 * CC];

  const int wave = threadIdx.x >> 5;
  const int lane = threadIdx.x & 31;
  const int tile = blockIdx.x * 8 + wave;
  const int row0 = tile * 16;
  const bool active = row0 < N;  // wave-uniform

  // stage W into LDS once per block
  for (int t = threadIdx.x; t < 2 * CC; t += 256) ldsW[t] = W[t];

  if (active) {
    // prefetch next block's tile region into L2
    __builtin_prefetch(x + (size_t)(row0 + 128) * CC, 0, 1);

    const unsigned ldsOff = (unsigned)(uintptr_t)(&ldsX[wave][0]);
    const u64 ga = (u64)(uintptr_t)(x + (size_t)row0 * CC);
    const unsigned rowsLeft = (unsigned)(N - row0);

#if __has_builtin(__builtin_amdgcn_tensor_load_to_lds)
    // ---- Tensor DMA descriptor (D#), groups 0..3 ----
    v4u g0;
    g0[0] = 1u;                                          // count=1, user mode
    g0[1] = ldsOff;                                      // lds_addr (bytes)
    g0[2] = (unsigned)(ga & 0xFFFFFFFFull);              // global_addr[31:0]
    g0[3] = (unsigned)((ga >> 32) & 0x1FFFFFFull)        // global_addr[56:32]
            | (2u << 30);                                // type=2 (image)
    v8i g1;
    g1[0] = (int)((2u << 16)                             // data_size=2 -> 4B
                  | (1u << 20)                           // pad_enable
                  | (6u << 22)                           // pad_interval: 128 dwords
                  | (0u << 25));                         // pad_amount: 1 dword
    g1[1] = (int)(((unsigned)CC & 0xFFFFu) << 16);       // tensor_dim0[15:0]
    g1[2] = (int)(((rowsLeft & 0xFFFFu) << 16)           // tensor_dim1[15:0]
                  | ((unsigned)CC >> 16));               // tensor_dim0[31:16]
    g1[3] = (int)((((unsigned)CC) << 16)                 // tile_dim0 = C
                  | ((rowsLeft >> 16) & 0xFFFFu));       // tensor_dim1[31:16]
    g1[4] = 16;                                          // tile_dim1=16, tile_dim2=0
    g1[5] = CC;                                          // tensor_dim0_stride[31:0]
    g1[6] = 0;
    g1[7] = 0;
    v4i g2 = {0, 0, 0, 0};
    v4i g3 = {0, 0, 0, 0};
    v8i g4 = {0, 0, 0, 0, 0, 0, 0, 0};
    // 6-arg toolchain variant: (g0, g1, g2, g3, g4, cpol)
    __builtin_amdgcn_tensor_load_to_lds(g0, g1, g2, g3, g4, 0);
#else
    // fallback: manual strided copy into the same LDS layout
    for (int t = lane; t < 16 * CC; t += 32) {
      int r = t >> 7, c = t & (CC - 1);
      unsigned rr = (unsigned)r;
      ldsX[wave][r * CSTRIDE + c] =
          (rr < rowsLeft) ? x[(size_t)(row0 + r) * CC + c] : 0.0f;
    }
#endif
  }
  __syncthreads();  // ldsW visible to all waves

  if (active) {
#if __has_builtin(__builtin_amdgcn_tensor_load_to_lds)
    __builtin_amdgcn_s_wait_tensorcnt(0);  // own tile landed in LDS
#endif
    const int m = lane & 15;
    const int kh = (lane >> 4) << 1;  // 0 or 2
    const float* xw = &ldsX[wave][m * CSTRIDE];
    const int wbase = (m == 1) ? CC : 0;
    const float msk = (m < 2) ? 1.0f : 0.0f;

    v8f acc = {};
#pragma unroll 4
    for (int k0 = 0; k0 < CC; k0 += 4) {
      const int kb = k0 + kh;
      v2f a, b;
      a.x = xw[kb + 0];               // ds_load_b64, no divergence
      a.y = xw[kb + 1];
      b.x = ldsW[wbase + kb + 0] * msk;
      b.y = ldsW[wbase + kb + 1] * msk;
      acc = __builtin_amdgcn_wmma_f32_16x16x4_f32(false, a, false, b,
                                                  (short)0, acc, false, false);
    }
    if (m < 2) {  // lanes 0,16 -> p1 ; lanes 1,17 -> p2
      float* dp = (m == 0) ? p1 : p2;
      const int mb = row0 + ((lane >> 4) << 3);
#pragma unroll
      for (int r = 0; r < 8; ++r) {
        const int rr = mb + r;
        if (rr < N) dp[rr] = acc[r];
      }
    }
  }
}

// ---------- edge scoring + segment softmax ----------
__global__ void k_edge_raw(const int* __restrict__ src, const int* __restrict__ dst,
                           const float* __restrict__ p1, const float* __restrict__ p2,
                           const float* __restrict__ bptr, float* __restrict__ raw,
                           unsigned* __restrict__ mEnc, int E) {
  int e = blockIdx.x * blockDim.x + threadIdx.x;
  if (e >= E) return;
  float r = p1[src[e]] + p2[dst[e]] + bptr[0];
  raw[e] = r;
  atomicMax(&mEnc[dst[e]], f32_ord(r));
}
__global__ void k_edge_exp(const int* __restrict__ dst, const float* __restrict__ raw,
                           const unsigned* __restrict__ mEnc, float* __restrict__ ex,
                           float* __restrict__ den, int E) {
  int e = blockIdx.x * blockDim.x + threadIdx.x;
  if (e >= E) return;
  int t = dst[e];
  float v = expf(raw[e] - ord_f32(mEnc[t]));
  ex[e] = v;
  atomicAdd(&den[t], v);
}
__global__ void k_edge_score(const int* __restrict__ dst, const float* __restrict__ ex,
                             const float* __restrict__ den, float* __restrict__ score, int E) {
  int e = blockIdx.x * blockDim.x + threadIdx.x;
  if (e >= E) return;
  score[e] = ex[e] / (den[dst[e]] + 1e-16f) + 0.5f;
}

// ---------- hybrid LDS/global bitonic sort on u64 keys ----------
__global__ void k_sort_init_score(const float* __restrict__ score, u64* __restrict__ key,
                                  int E, int S) {
  int i = blockIdx.x * blockDim.x + threadIdx.x;
  if (i >= S) return;
  // stable descending: tie-break by ascending index
  key[i] = (i < E) ? (((u64)f32_ord(score[i]) << 32) | (u64)(0xFFFFFFFFu - (unsigned)i))
                   : 0ull;  // padding sorts last (descending)
}
// all kk<=1024 stages in LDS: builds alternating 1024-sorted runs
__global__ __launch_bounds__(256) void k_bitonic_lds_presort(u64* __restrict__ key, int desc) {
  __shared__ u64 sh[1024];
  const int base = blockIdx.x * 1024;
  for (int t = threadIdx.x; t < 1024; t += 256) sh[t] = key[base + t];
  __syncthreads();
  for (int kk = 2; kk <= 1024; kk <<= 1) {
    for (int j = kk >> 1; j > 0; j >>= 1) {
#pragma unroll
      for (int q = 0; q < 2; ++q) {
        int c = threadIdx.x + q * 256;  // 512 disjoint comparators per stage
        int i = ((c & ~(j - 1)) << 1) | (c & (j - 1));
        int gi = base + i;
        bool bigFirst = (((gi & kk) == 0) == (desc != 0));
        u64 a = sh[i], b = sh[i + j];
        if (bigFirst ? (a < b) : (a > b)) { sh[i] = b; sh[i + j] = a; }
      }
      __syncthreads();
    }
  }
  for (int t = threadIdx.x; t < 1024; t += 256) key[base + t] = sh[t];
}
// j<=512 tail of a merge stage kk, in LDS
__global__ __launch_bounds__(256) void k_bitonic_lds_merge(u64* __restrict__ key,
                                                           int kk, int desc) {
  __shared__ u64 sh[1024];
  const int base = blockIdx.x * 1024;
  for (int t = threadIdx.x; t < 1024; t += 256) sh[t] = key[base + t];
  __syncthreads();
  for (int j = 512; j > 0; j >>= 1) {
#pragma unroll
    for (int q = 0; q < 2; ++q) {
      int c = threadIdx.x + q * 256;
      int i = ((c & ~(j - 1)) << 1) | (c & (j - 1));
      int gi = base + i;
      bool bigFirst = (((gi & kk) == 0) == (desc != 0));
      u64 a = sh[i], b = sh[i + j];
      if (bigFirst ? (a < b) : (a > b)) { sh[i] = b; sh[i + j] = a; }
    }
    __syncthreads();
  }
  for (int t = threadIdx.x; t < 1024; t += 256) key[base + t] = sh[t];
}
// global stage for j >= 1024
__global__ void k_bitonic(u64* __restrict__ key, int n, int j, int kk, int desc) {
  int i = blockIdx.x * blockDim.x + threadIdx.x;
  if (i >= n) return;
  int ixj = i ^ j;
  if (ixj > i) {
    u64 a = key[i], b = key[ixj];
    bool bigFirst = (((i & kk) == 0) == (desc != 0));
    bool sw = bigFirst ? (a < b) : (a > b);
    if (sw) { key[i] = b; key[ixj] = a; }
  }
}

// ---------- sequential greedy contraction (exact reference semantics) ----------
__global__ void k_greedy(const int* __restrict__ src, const int* __restrict__ dst,
                         const u64* __restrict__ skey, int* __restrict__ mask,
                         int* __restrict__ cluster, float* __restrict__ spc,
                         int* __restrict__ total, int E, int N) {
  if (threadIdx.x != 0 || blockIdx.x != 0) return;
  int cnt = 0;
  for (int i = 0; i < E; ++i) {
    u64 k = skey[i];
    int e = (int)(0xFFFFFFFFu - (unsigned)(k & 0xFFFFFFFFull));
    int s = src[e], t = dst[e];
    if (mask[s] && mask[t]) {
      cluster[s] = cnt;
      cluster[t] = cnt;
      mask[s] = 0;
      mask[t] = 0;
      spc[cnt] = ord_f32((unsigned)(k >> 32));
      ++cnt;
    }
  }
  int c = cnt;
  for (int n = 0; n < N; ++n)
    if (mask[n]) cluster[n] = c++;
  *total = c;
}

// ---------- new_x = segment_sum(x, cluster) * scores_pc ----------
__global__ void k_scatter_x(const float* __restrict__ x, const int* __restrict__ cluster,
                            float* __restrict__ newx, int N, int C) {
  int i = blockIdx.x * blockDim.x + threadIdx.x;
  if (i >= N * C) return;
  int n = i / C, c = i - n * C;
  atomicAdd(&newx[(size_t)cluster[n] * C + c], x[i]);
}
__global__ void k_scale_x(float* __restrict__ newx, const float* __restrict__ spc,
                          int N, int C) {
  int i = blockIdx.x * blockDim.x + threadIdx.x;
  if (i >= N * C) return;
  newx[i] *= spc[i / C];
}
__global__ void k_scatter_batch(const int* __restrict__ batch, const int* __restrict__ cluster,
                                int* __restrict__ nb, int N) {
  int i = blockIdx.x * blockDim.x + threadIdx.x;
  if (i < N) nb[cluster[i]] = batch[i];
}

// ---------- unique(cluster[edge_index]) via sort + flags + scan + compact ----------
__global__ void k_make_keys(const int* __restrict__ src, const int* __restrict__ dst,
                            const int* __restrict__ cluster, u64* __restrict__ key,
                            int E, int S) {
  int i = blockIdx.x * blockDim.x + threadIdx.x;
  if (i >= S) return;
  key[i] = (i < E) ? (((u64)(unsigned)cluster[src[i]] << 32) | (u64)(unsigned)cluster[dst[i]])
                   : ~0ull;  // padding sorts last (ascending)
}
__global__ void k_flags(const u64* __restrict__ key, int* __restrict__ flags, int S) {
  int i = blockIdx.x * blockDim.x + threadIdx.x;
  if (i >= S) return;
  u64 k = key[i];
  flags[i] = (k != ~0ull) && (i == 0 || k != key[i - 1]);
}
__global__ void k_scan_block(const int* __restrict__ flags, int* __restrict__ pos,
                             int* __restrict__ bsums, int S) {
  __shared__ int sh[1024];
  const int base = blockIdx.x * 1024;
  for (int t = threadIdx.x; t < 1024; t += 256) sh[t] = flags[base + t];
  __syncthreads();
  for (int off = 1; off < 1024; off <<= 1) {
    int vals[4];
#pragma unroll
    for (int q = 0; q < 4; ++q) {
      int t = threadIdx.x + q * 256;
      vals[q] = (t >= off) ? sh[t - off] : 0;
    }
    __syncthreads();
#pragma unroll
    for (int q = 0; q < 4; ++q) sh[threadIdx.x + q * 256] += vals[q];
    __syncthreads();
  }
  for (int t = threadIdx.x; t < 1024; t += 256) pos[base + t] = sh[t] - flags[base + t];
  if (threadIdx.x == 0) bsums[blockIdx.x] = sh[1023];
}
__global__ void k_scan_sums(int* __restrict__ bsums, int nb, int* __restrict__ nuniq) {
  __shared__ int sh[1024];
  int orig[4];
#pragma unroll
  for (int q = 0; q < 4; ++q) {
    int t = threadIdx.x + q * 256;
    int v = (t < nb) ? bsums[t] : 0;
    orig[q] = v;
    sh[t] = v;
  }
  __syncthreads();
  for (int off = 1; off < 1024; off <<= 1) {
    int vals[4];
#pragma unroll
    for (int q = 0; q < 4; ++q) {
      int t = threadIdx.x + q * 256;
      vals[q] = (t >= off) ? sh[t - off] : 0;
    }
    __syncthreads();
#pragma unroll
    for (int q = 0; q < 4; ++q) sh[threadIdx.x + q * 256] += vals[q];
    __syncthreads();
  }
#pragma unroll
  for (int q = 0; q < 4; ++q) {
    int t = threadIdx.x + q * 256;
    if (t < nb) bsums[t] = sh[t] - orig[q];
  }
  if (threadIdx.x == 0) *nuniq = sh[1023];
}
__global__ void k_scan_add(int* __restrict__ pos, const int* __restrict__ bsums, int S) {
  int i = blockIdx.x * blockDim.x + threadIdx.x;
  if (i < S) pos[i] += bsums[i >> 10];
}
__global__ void k_scatter_unique(const u64* __restrict__ key, const int* __restrict__ flags,
                                 const int* __restrict__ pos, int* __restrict__ ei,
                                 int S, int E) {
  int i = blockIdx.x * blockDim.x + threadIdx.x;
  if (i >= S || !flags[i]) return;
  int p = pos[i];
  if (p < E) {
    u64 k = key[i];
    ei[p]     = (int)(unsigned)(k >> 32);
    ei[E + p] = (int)(unsigned)(k & 0xFFFFFFFFull);
  }
}
__global__ void k_pad_edges(int* __restrict__ ei, const int* __restrict__ nuniq, int E, int N) {
  int i = blockIdx.x * blockDim.x + threadIdx.x;
  if (i >= E) return;
  if (i >= *nuniq) { ei[i] = N; ei[E + i] = N; }
}

// ---------- launch ----------
extern "C" void kernel_launch(void* const* d_in, const int* in_sizes, int n_in,
                              void* d_out, int out_size, void* d_ws, size_t ws_size,
                              hipStream_t stream) {
  const float* x     = (const float*)d_in[0];
  const int*   eidx  = (const int*)d_in[1];
  const int*   batch = (const int*)d_in[2];
  const float* W     = (const float*)d_in[3];
  const float* b     = (const float*)d_in[4];

  const int N = in_sizes[2];
  const int E = in_sizes[1] / 2;
  const int C = in_sizes[3] / 2;  // == CC (128)
  const int* src = eidx;
  const int* dst = eidx + E;

  // output layout (flat, reference return order)
  float* out_newx  = (float*)d_out;                     // N*C
  int*   out_ei    = (int*)(out_newx + (size_t)N * C);  // 2*E
  int*   out_nb    = out_ei + 2 * (size_t)E;            // N
  int*   out_cl    = out_nb + N;                        // N
  float* out_spc   = (float*)(out_cl + N);              // N
  int*   out_total = (int*)(out_spc + N);               // 1

  // workspace bump allocator
  char* w = (char*)d_ws;
  auto alloc = [&](size_t bytes) -> void* {
    void* p = (void*)w;
    w += (bytes + 255) & ~(size_t)255;
    return p;
  };
  int S = 1024;
  while (S < E) S <<= 1;  // 1<<20 for E=800000

  float*    p1    = (float*)alloc((size_t)N * 4);
  float*    p2    = (float*)alloc((size_t)N * 4);
  float*    raw   = (float*)alloc((size_t)E * 4);
  float*    ex    = (float*)alloc((size_t)E * 4);
  float*    score = (float*)alloc((size_t)E * 4);
  unsigned* mEnc  = (unsigned*)alloc((size_t)N * 4);
  float*    den   = (float*)alloc((size_t)N * 4);
  u64*      keyA  = (u64*)alloc((size_t)S * 8);  // score sort
  u64*      keyB  = (u64*)alloc((size_t)S * 8);  // unique-edge sort
  int*      mask  = (int*)alloc((size_t)N * 4);
  int*      flags = (int*)alloc((size_t)S * 4);
  int*      pos   = (int*)alloc((size_t)S * 4);
  int*      bsums = (int*)alloc((size_t)(S / 1024) * 4);
  int*      nuniq = (int*)alloc(4);

  auto gr = [](int n) { return (n + THREADS - 1) / THREADS; };
  auto run_sort = [&](u64* key, int desc) {
    k_bitonic_lds_presort<<<S / 1024, 256, 0, stream>>>(key, desc);
    for (int kk = 2048; kk <= S; kk <<= 1) {
      for (int j = kk >> 1; j >= 1024; j >>= 1)
        k_bitonic<<<gr(S), THREADS, 0, stream>>>(key, S, j, kk, desc);
      k_bitonic_lds_merge<<<S / 1024, 256, 0, stream>>>(key, kk, desc);
    }
  };

  // 1) node projections: TDM -> LDS -> WMMA f32 16x16x4
  {
    int tiles = (N + 15) / 16;
    int blocks = (tiles + 7) / 8;
    k_proj<<<blocks, 256, 0, stream>>>(x, W, p1, p2, N);
  }

  // 2) edge raw scores + segment softmax + 0.5
  k_fill_i32<<<gr(N), THREADS, 0, stream>>>((int*)mEnc, 0, N);
  k_fill_f32<<<gr(N), THREADS, 0, stream>>>(den, 0.0f, N);
  k_edge_raw<<<gr(E), THREADS, 0, stream>>>(src, dst, p1, p2, b, raw, mEnc, E);
  k_edge_exp<<<gr(E), THREADS, 0, stream>>>(dst, raw, mEnc, ex, den, E);
  k_edge_score<<<gr(E), THREADS, 0, stream>>>(dst, ex, den, score, E);

  // 3) stable descending sort of scores
  k_sort_init_score<<<gr(S), THREADS, 0, stream>>>(score, keyA, E, S);
  run_sort(keyA, /*desc=*/1);

  // 4) sequential greedy contraction + singleton assignment + scores_pc
  k_fill_i32<<<gr(N), THREADS, 0, stream>>>(mask, 1, N);
  k_fill_i32<<<gr(N), THREADS, 0, stream>>>(out_cl, 0, N);
  k_fill_f32<<<gr(N), THREADS, 0, stream>>>(out_spc, 1.0f, N);
  k_greedy<<<1, 1, 0, stream>>>(src, dst, keyA, mask, out_cl, out_spc, out_total, E, N);

  // 5) new_x = segment_sum(x, cluster) * scores_pc
  k_fill_f32<<<gr(N * C), THREADS, 0, stream>>>(out_newx, 0.0f, N * C);
  k_scatter_x<<<gr(N * C), THREADS, 0, stream>>>(x, out_cl, out_newx, N, C);
  k_scale_x<<<gr(N * C), THREADS, 0, stream>>>(out_newx, out_spc, N, C);

  // 6) new_batch
  k_fill_i32<<<gr(N), THREADS, 0, stream>>>(out_nb, 0, N);
  k_scatter_batch<<<gr(N), THREADS, 0, stream>>>(batch, out_cl, out_nb, N);

  // 7) new_edge_index = unique(cluster[edge_index], sorted, padded with N)
  k_make_keys<<<gr(S), THREADS, 0, stream>>>(src, dst, out_cl, keyB, E, S);
  run_sort(keyB, /*desc=*/0);
  k_flags<<<gr(S), THREADS, 0, stream>>>(keyB, flags, S);
  k_scan_block<<<S / 1024, 256, 0, stream>>>(flags, pos, bsums, S);
  k_scan_sums<<<1, 256, 0, stream>>>(bsums, S / 1024, nuniq);
  k_scan_add<<<gr(S), THREADS, 0, stream>>>(pos, bsums, S);
  k_scatter_unique<<<gr(S), THREADS, 0, stream>>>(keyB, flags, pos, out_ei, S, E);
  k_pad_edges<<<gr(E), THREADS, 0, stream>>>(out_ei, nuniq, E, N);
}